// VecInt_2688649527744
// MI455X (gfx1250) — compile-verified
//
#include <hip/hip_runtime.h>

// VecInt (scaling-and-squaring diffeomorphic integration), MI455X / gfx1250.
//
// Memory-bound gather workload: 7 passes of v <- v + trilinear(v, grid+v).
// Traffic floor ~1.7 GB -> ~73us at 23.3 TB/s HBM. No GEMM content -> WMMA
// is inapplicable; optimization targets are pass fusion, L2 residency,
// division-free 3D indexing, coalesced b96 vmem, NT hints on the final
// write, and WGP-scope prefetch of the likely gather plane.

namespace {

constexpr int Dd = 160, Hh = 192, Ww = 160;
constexpr int NVOX  = Dd * Hh * Ww;   // 4,915,200 voxels per batch
constexpr int NEL   = NVOX * 3;       // 14,745,600 floats per batch
constexpr float SSCALE = 1.0f / 128.0f;  // OUT_TIME_PT / 2^INT_STEPS

// One squaring step: vout[p] = vin[p] + trilinear(vin, p + vin[p]).
// FIRST: vin is the raw velocity; apply the 1/128 scale on the fly
//        (lerp is linear in the sampled values, so tri(s*v) = s*tri(v)).
// LAST : vout is the kernel's final output; store non-temporally.
//
// Launch: block (160, 2, 1), grid (96, 160, batch) -> every thread maps 1:1
// to a voxel with zero integer divisions and no bounds checks.
template <bool FIRST, bool LAST>
__global__ __launch_bounds__(320) void vecint_step(
    const float* __restrict__ vin, float* __restrict__ vout)
{
    const int z = threadIdx.x;                        // 0..159 (W, contiguous)
    const int y = blockIdx.x * 2 + threadIdx.y;       // 0..191 (H)
    const int x = blockIdx.y;                         // 0..159 (D)
    const float* __restrict__ v = vin  + (long)blockIdx.z * NEL;
    float*       __restrict__ o = vout + (long)blockIdx.z * NEL;

    const int r    = (x * Hh + y) * Ww + z;
    const int base = r * 3;

    // The gather below lands within ~3 voxels of (x,y,z); the x+1 plane at the
    // same (y,z) is the one line not already being streamed by this block.
    // WGP-scope prefetch (locality 3) overlaps the displacement load.
    __builtin_prefetch(v + base + Hh * Ww * 3, 0, 3);

    const float rx = v[base + 0];
    const float ry = v[base + 1];
    const float rz = v[base + 2];
    const float dx = FIRST ? rx * SSCALE : rx;
    const float dy = FIRST ? ry * SSCALE : ry;
    const float dz = FIRST ? rz * SSCALE : rz;

    // Sample location, clamped to the volume (matches jnp.clip then floor).
    const float lx = fminf(fmaxf((float)x + dx, 0.0f), (float)(Dd - 1));
    const float ly = fminf(fmaxf((float)y + dy, 0.0f), (float)(Hh - 1));
    const float lz = fminf(fmaxf((float)z + dz, 0.0f), (float)(Ww - 1));

    const float fx = floorf(lx), fy = floorf(ly), fz = floorf(lz);
    const float wx = lx - fx,   wy = ly - fy,   wz = lz - fz;
    const int x0 = (int)fx, y0 = (int)fy, z0 = (int)fz;
    const int x1 = min(x0 + 1, Dd - 1);
    const int y1 = min(y0 + 1, Hh - 1);
    const int z1 = min(z0 + 1, Ww - 1);

    // Row bases for the 4 (x,y) corner rows; 8 corner gathers as b96 loads.
    const int r00 = (x0 * Hh + y0) * Ww;
    const int r01 = (x0 * Hh + y1) * Ww;
    const int r10 = (x1 * Hh + y0) * Ww;
    const int r11 = (x1 * Hh + y1) * Ww;
    const int a000 = (r00 + z0) * 3, a001 = (r00 + z1) * 3;
    const int a010 = (r01 + z0) * 3, a011 = (r01 + z1) * 3;
    const int a100 = (r10 + z0) * 3, a101 = (r10 + z1) * 3;
    const int a110 = (r11 + z0) * 3, a111 = (r11 + z1) * 3;

    const float omz = 1.0f - wz, omy = 1.0f - wy, omx = 1.0f - wx;

    float res[3];
#pragma unroll
    for (int c = 0; c < 3; ++c) {
        const float c000 = v[a000 + c], c001 = v[a001 + c];
        const float c010 = v[a010 + c], c011 = v[a011 + c];
        const float c100 = v[a100 + c], c101 = v[a101 + c];
        const float c110 = v[a110 + c], c111 = v[a111 + c];
        const float a00 = c000 * omz + c001 * wz;
        const float a01 = c010 * omz + c011 * wz;
        const float a10 = c100 * omz + c101 * wz;
        const float a11 = c110 * omz + c111 * wz;
        const float b0  = a00 * omy + a01 * wy;
        const float b1  = a10 * omy + a11 * wy;
        res[c] = b0 * omx + b1 * wx;
    }

    float o0, o1, o2;
    if (FIRST) {
        // out = s*(raw + tri(raw at scaled locs))  ==  v0 + tri(v0, p+v0)
        o0 = SSCALE * (rx + res[0]);
        o1 = SSCALE * (ry + res[1]);
        o2 = SSCALE * (rz + res[2]);
    } else {
        o0 = rx + res[0];
        o1 = ry + res[1];
        o2 = rz + res[2];
    }

    if (LAST) {
        // Final output is never re-read by this launch chain: NT store so the
        // 118 MB write does not rinse the L2-resident field.
        __builtin_nontemporal_store(o0, &o[base + 0]);
        __builtin_nontemporal_store(o1, &o[base + 1]);
        __builtin_nontemporal_store(o2, &o[base + 2]);
    } else {
        o[base + 0] = o0;
        o[base + 1] = o1;
        o[base + 2] = o2;
    }
}

}  // namespace

extern "C" void kernel_launch(void* const* d_in, const int* in_sizes, int n_in,
                              void* d_out, int out_size, void* d_ws, size_t ws_size,
                              hipStream_t stream)
{
    (void)in_sizes; (void)n_in; (void)out_size; (void)ws_size;

    const float* vel  = (const float*)d_in[0];
    float* bufO = (float*)d_out;  // ping
    float* bufW = (float*)d_ws;   // pong: needs 2*NEL*4 = 117,964,800 bytes

    const dim3 block(Ww, 2, 1);            // 320 threads = 10 waves, exact row
    const dim3 grid(Hh / 2, Dd, 2);        // (96, 160, batch)

    // Step 0 fuses the 1/128 scale (saves a full 236 MB pass): vel -> d_out.
    // Parity: O,W,O,W,O,W,O -- 7 writes, last one lands in d_out.
    vecint_step<true, false><<<grid, block, 0, stream>>>(vel, bufO);

    const float* cur = bufO;
    float*       nxt = bufW;
    for (int s = 1; s < 7; ++s) {
        if (s == 6)
            vecint_step<false, true ><<<grid, block, 0, stream>>>(cur, nxt);
        else
            vecint_step<false, false><<<grid, block, 0, stream>>>(cur, nxt);
        float* tmp = (float*)cur; cur = nxt; nxt = tmp;
    }
}